// MultiHeadAttention_62285615727132
// MI455X (gfx1250) — compile-verified
//
#include <hip/hip_runtime.h>
#include <hip/hip_bf16.h>

// MHA forward for MI455X (gfx1250, wave32). fp32 end-to-end via
// V_WMMA_F32_16X16X4_F32 (full precision; kernel is HBM-bound on the 537MB
// attn output, so the fp32 matrix path costs nothing vs bf16).
// Tile staging uses CDNA5 async global->LDS movers (ASYNCcnt) to skip the
// VGPR round trip.

#define B_ 2
#define S_ 2048
#define D_ 1024
#define H_ 16
#define HD_ 64

typedef __attribute__((ext_vector_type(2))) float v2f;
typedef __attribute__((ext_vector_type(4))) float v4f;
typedef __attribute__((ext_vector_type(8))) float v8f;

static __device__ __forceinline__ v8f wmma_f32(v2f a, v2f b, v8f c) {
  // D = A(16x4) * B(4x16) + C(16x16), fp32
  return __builtin_amdgcn_wmma_f32_16x16x4_f32(
      /*neg_a=*/false, a, /*neg_b=*/false, b,
      /*c_mod=*/(short)0, c, /*reuse_a=*/false, /*reuse_b=*/false);
}

// Async global->LDS 16-byte mover (GLOBAL_LOAD_ASYNC_TO_LDS_B128, ASYNCcnt).
// LDS destination = low 32 bits of the generic shared-memory pointer.
static __device__ __forceinline__ void async_g2l_b128(const void* gptr, void* lptr) {
  unsigned lofs = (unsigned)(uintptr_t)lptr;
  asm volatile("global_load_async_to_lds_b128 %0, %1, off"
               :: "v"(lofs), "v"(gptr)
               : "memory");
}
static __device__ __forceinline__ void wait_async0() {
  asm volatile("s_wait_asynccnt 0" ::: "memory");
}

// ---------------------------------------------------------------------------
// GEMM: Y[M,N] = X[M,K] @ W[K,N] + bias[N]
// Block tile 64x128, 256 threads = 8 waves, wave tile 32x32 (2x2 WMMA f32).
// ---------------------------------------------------------------------------
#define TM 64
#define TN 128
#define TK 32

__global__ __launch_bounds__(256) void gemm_bias_wmma(
    const float* __restrict__ X, const float* __restrict__ W,
    const float* __restrict__ bias, float* __restrict__ Y,
    int M, int N, int K) {
  __shared__ float As[TM][TK + 4];   // stride 36 -> conflict-free column reads
  __shared__ float Bs[TK][TN + 4];   // stride 132

  const int tid  = threadIdx.x;
  const int lane = tid & 31;
  const int wave = tid >> 5;
  const int wm = wave & 1;           // 2 wave-rows
  const int wn = wave >> 1;          // 4 wave-cols
  const int lo = lane & 15;
  const int hi = lane >> 4;
  const int block_m = blockIdx.y * TM;
  const int block_n = blockIdx.x * TN;

  v8f acc[2][2] = {};

  for (int k0 = 0; k0 < K; k0 += TK) {
    // Stage A tile 64x32: 512 x float4, 2 async movers per thread.
    {
      const int r = tid >> 3;            // 0..31
      const int c = (tid & 7) << 2;      // 0..28
#pragma unroll
      for (int i = 0; i < 2; ++i) {
        const int row = r + i * 32;
        async_g2l_b128(&X[(size_t)(block_m + row) * K + k0 + c], &As[row][c]);
      }
    }
    // Stage B tile 32x128: 1024 x float4, 4 async movers per thread.
    {
      const int r = tid >> 5;            // 0..7
      const int c = (tid & 31) << 2;     // 0..124
#pragma unroll
      for (int i = 0; i < 4; ++i) {
        const int row = r + i * 8;
        async_g2l_b128(&W[(size_t)(k0 + row) * N + block_n + c], &Bs[row][c]);
      }
    }
    wait_async0();                       // LDS writes resident for this wave
    __syncthreads();                     // visible workgroup-wide

    // Prefetch next K-chunk into L2 (global_prefetch_b8).
    if (k0 + TK < K) {
      __builtin_prefetch(&X[(size_t)(block_m + (tid >> 3)) * K + k0 + TK], 0, 1);
      __builtin_prefetch(&W[(size_t)(k0 + TK + (tid >> 5)) * N + block_n + ((tid & 31) << 2)], 0, 1);
    }

#pragma unroll
    for (int k4 = 0; k4 < TK / 4; ++k4) {
      const int kk = k4 * 4 + hi * 2;    // lane half picks K pair {0,1}/{2,3}
      v2f a[2], b[2];
#pragma unroll
      for (int mi = 0; mi < 2; ++mi) {
        const int m = wm * 32 + mi * 16 + lo;
        a[mi] = (v2f){As[m][kk], As[m][kk + 1]};
      }
#pragma unroll
      for (int ni = 0; ni < 2; ++ni) {
        const int n = wn * 32 + ni * 16 + lo;
        b[ni] = (v2f){Bs[kk][n], Bs[kk + 1][n]};
      }
#pragma unroll
      for (int mi = 0; mi < 2; ++mi)
#pragma unroll
        for (int ni = 0; ni < 2; ++ni)
          acc[mi][ni] = wmma_f32(a[mi], b[ni], acc[mi][ni]);
    }
    __syncthreads();
  }

  // Epilogue: C/D layout -> VGPR v holds row (v + 8*hi), col = lo.
#pragma unroll
  for (int mi = 0; mi < 2; ++mi)
#pragma unroll
    for (int ni = 0; ni < 2; ++ni) {
      const int col = block_n + wn * 32 + ni * 16 + lo;
      const float bv = bias[col];
#pragma unroll
      for (int v = 0; v < 8; ++v) {
        const int row = block_m + wm * 32 + mi * 16 + v + 8 * hi;
        Y[(size_t)row * N + col] = acc[mi][ni][v] + bv;
      }
    }
}

// ---------------------------------------------------------------------------
// Fused attention: scores = scale * Q Kᵀ, softmax, attn written once to d_out,
// context = attn @ V accumulated on-chip. One block = 16 q-rows of one head;
// 4 waves each own a 512-column strip of the 2048 keys.
// ---------------------------------------------------------------------------
#define QB 16

__global__ __launch_bounds__(128) void attn_fused(
    const float* __restrict__ Q, const float* __restrict__ Kt,
    const float* __restrict__ V, float* __restrict__ attn,
    float* __restrict__ ctx) {
  __shared__ float Qs[QB][HD_ + 4];          // stride 68
  __shared__ float pbuf[4][QB][17];          // per-wave P restage (C/D -> A)
  __shared__ float wmx[4][QB], wsm[4][QB];
  __shared__ float mfin[QB], lrec[QB];
  __shared__ float ctxred[4][QB][HD_];

  const int tid  = threadIdx.x;
  const int lane = tid & 31;
  const int wave = tid >> 5;
  const int lo = lane & 15;
  const int hi = lane >> 4;

  const int q0 = blockIdx.x * QB;
  const int bh = blockIdx.y;
  const int b  = bh / H_;
  const int h  = bh % H_;
  const size_t head_base = (size_t)b * S_ * D_ + (size_t)h * HD_;
  const float scale = 0.125f;                // 1/sqrt(64)

  // Stage Q tile (16 x 64): 256 x float4, 2 async movers per thread.
#pragma unroll
  for (int i = 0; i < 2; ++i) {
    const int s4 = tid + i * 128;
    const int row = s4 >> 4;
    const int c = (s4 & 15) << 2;
    async_g2l_b128(&Q[head_base + (size_t)(q0 + row) * D_ + c], &Qs[row][c]);
  }
  wait_async0();
  __syncthreads();

  const int n_begin = wave * (S_ / 4);

  // ---- Pass 1: online row max / sum over this wave's 512-key strip ----
  float rmax[8], rsum[8];
#pragma unroll
  for (int v = 0; v < 8; ++v) { rmax[v] = -1e30f; rsum[v] = 0.0f; }

  for (int t = 0; t < (S_ / 4) / 16; ++t) {
    const int n0 = n_begin + t * 16;
    v8f sc = {};
#pragma unroll
    for (int k4 = 0; k4 < HD_ / 4; ++k4) {
      const int kk = k4 * 4 + hi * 2;
      v2f a = (v2f){Qs[lo][kk], Qs[lo][kk + 1]};
      const float* kr = &Kt[head_base + (size_t)(n0 + lo) * D_ + kk];
      v2f bf = (v2f){kr[0], kr[1]};
      sc = wmma_f32(a, bf, sc);
    }
#pragma unroll
    for (int v = 0; v < 8; ++v) {
      const float s = sc[v] * scale;
      float tmax = s;
      for (int off = 8; off; off >>= 1) tmax = fmaxf(tmax, __shfl_xor(tmax, off, 16));
      const float nm = fmaxf(rmax[v], tmax);
      float tsum = __expf(s - nm);
      for (int off = 8; off; off >>= 1) tsum += __shfl_xor(tsum, off, 16);
      rsum[v] = rsum[v] * __expf(rmax[v] - nm) + tsum;
      rmax[v] = nm;
    }
  }

  if (lo == 0) {
#pragma unroll
    for (int v = 0; v < 8; ++v) {
      wmx[wave][v + 8 * hi] = rmax[v];
      wsm[wave][v + 8 * hi] = rsum[v];
    }
  }
  __syncthreads();
  if (tid < QB) {
    float m = wmx[0][tid];
    for (int w = 1; w < 4; ++w) m = fmaxf(m, wmx[w][tid]);
    float l = 0.0f;
    for (int w = 0; w < 4; ++w) l += wsm[w][tid] * __expf(wmx[w][tid] - m);
    mfin[tid] = m;
    lrec[tid] = 1.0f / l;
  }
  __syncthreads();

  float Mreg[8], Lreg[8];
#pragma unroll
  for (int v = 0; v < 8; ++v) {
    Mreg[v] = mfin[v + 8 * hi];
    Lreg[v] = lrec[v + 8 * hi];
  }

  // ---- Pass 2: recompute scores, normalize, emit attn, accumulate PV ----
  v8f cacc[4] = {};
  const size_t attn_base = (size_t)bh * S_ * S_;

  for (int t = 0; t < (S_ / 4) / 16; ++t) {
    const int n0 = n_begin + t * 16;
    v8f sc = {};
#pragma unroll
    for (int k4 = 0; k4 < HD_ / 4; ++k4) {
      const int kk = k4 * 4 + hi * 2;
      v2f a = (v2f){Qs[lo][kk], Qs[lo][kk + 1]};
      const float* kr = &Kt[head_base + (size_t)(n0 + lo) * D_ + kk];
      v2f bf = (v2f){kr[0], kr[1]};
      sc = wmma_f32(a, bf, sc);
    }
#pragma unroll
    for (int v = 0; v < 8; ++v) {
      const float p = __expf(sc[v] * scale - Mreg[v]) * Lreg[v];
      const int row = v + 8 * hi;
      attn[attn_base + (size_t)(q0 + row) * S_ + n0 + lo] = p;  // coalesced
      pbuf[wave][row][lo] = p;                                  // restage for A-frag
    }
    // PV: A = P(16x16 slice), B = V rows n0..n0+15. LDS in-order per wave.
#pragma unroll
    for (int j = 0; j < 4; ++j) {
#pragma unroll
      for (int k4 = 0; k4 < 4; ++k4) {
        const int kk = k4 * 4 + hi * 2;
        v2f a = (v2f){pbuf[wave][lo][kk], pbuf[wave][lo][kk + 1]};
        const float b0 = V[head_base + (size_t)(n0 + kk) * D_ + j * 16 + lo];
        const float b1 = V[head_base + (size_t)(n0 + kk + 1) * D_ + j * 16 + lo];
        v2f bf = (v2f){b0, b1};
        cacc[j] = wmma_f32(a, bf, cacc[j]);
      }
    }
  }

  // ---- Cross-wave context reduction ----
#pragma unroll
  for (int j = 0; j < 4; ++j)
#pragma unroll
    for (int v = 0; v < 8; ++v)
      ctxred[wave][v + 8 * hi][j * 16 + lo] = cacc[j][v];
  __syncthreads();

  for (int i = tid; i < QB * HD_; i += 128) {
    const int row = i >> 6;
    const int hd = i & 63;
    const float s = ctxred[0][row][hd] + ctxred[1][row][hd] +
                    ctxred[2][row][hd] + ctxred[3][row][hd];
    ctx[(size_t)b * S_ * D_ + (size_t)(q0 + row) * D_ + h * HD_ + hd] = s;
  }
}

// ---------------------------------------------------------------------------
extern "C" void kernel_launch(void* const* d_in, const int* in_sizes, int n_in,
                              void* d_out, int out_size, void* d_ws, size_t ws_size,
                              hipStream_t stream) {
  (void)in_sizes; (void)n_in; (void)out_size; (void)ws_size;

  const float* q  = (const float*)d_in[0];
  const float* k  = (const float*)d_in[1];
  const float* v  = (const float*)d_in[2];
  // d_in[3] = attn_mask: no effect in reference math (non-in-place masked_fill)
  const float* Wq = (const float*)d_in[4];
  const float* bq = (const float*)d_in[5];
  const float* Wk = (const float*)d_in[6];
  const float* bk = (const float*)d_in[7];
  const float* Wv = (const float*)d_in[8];
  const float* bv = (const float*)d_in[9];
  const float* Wo = (const float*)d_in[10];
  const float* bo = (const float*)d_in[11];

  float* out  = (float*)d_out;                           // [B,S,D]
  float* attn = (float*)d_out + (size_t)B_ * S_ * D_;    // [B,H,S,S]

  const size_t BSD = (size_t)B_ * S_ * D_;
  float* Qw  = (float*)d_ws;
  float* Kw  = Qw + BSD;
  float* Vw  = Kw + BSD;
  float* Ctx = Vw + BSD;

  const int M = B_ * S_;                                 // 4096
  dim3 gGemm(D_ / TN, M / TM);                           // (8, 64)
  dim3 bGemm(256);

  gemm_bias_wmma<<<gGemm, bGemm, 0, stream>>>(q, Wq, bq, Qw, M, D_, D_);
  gemm_bias_wmma<<<gGemm, bGemm, 0, stream>>>(k, Wk, bk, Kw, M, D_, D_);
  gemm_bias_wmma<<<gGemm, bGemm, 0, stream>>>(v, Wv, bv, Vw, M, D_, D_);

  dim3 gAttn(S_ / QB, B_ * H_);                          // (128, 32)
  attn_fused<<<gAttn, dim3(128), 0, stream>>>(Qw, Kw, Vw, attn, Ctx);

  gemm_bias_wmma<<<gGemm, bGemm, 0, stream>>>(Ctx, Wo, bo, out, M, D_, D_);
}